// TrajectoryAnalyzer_38070590111930
// MI455X (gfx1250) — compile-verified
//
#include <hip/hip_runtime.h>
#include <math.h>

// ---------------------------------------------------------------------------
// TrajectoryAnalyzer for MI455X (gfx1250, wave32)
//   in: features[N,256] f32, subject_ids[N] i64, baselines[64,256] f32,
//       W1[1,32], b1[32], W2[32,3], b2[3] (all f32)
//   out: logits[N,3] | dev[N] | scores[N]  (concatenated f32)
// Memory-bound: 512MB feature stream dominates (~23us floor @ 23.3TB/s).
// ---------------------------------------------------------------------------

typedef float f4  __attribute__((ext_vector_type(4)));
typedef float v2f __attribute__((ext_vector_type(2)));
typedef float v8f __attribute__((ext_vector_type(8)));

#define NSUB 64
#define DQ   64   // float4 chunks per 256-float row

// ---------------------------------------------------------------------------
// Kernel 1: per-row L2 deviation.  One wave per row: 2x b128 NT loads per lane
// stream the row; baselines live in LDS (64KB, loaded once per block).
// ---------------------------------------------------------------------------
__global__ __launch_bounds__(256) void dev_norm_kernel(
    const float* __restrict__ feat, const long long* __restrict__ sid,
    const float* __restrict__ base, float* __restrict__ raw,
    unsigned* __restrict__ maxslot, int nrows)
{
    __shared__ f4 sB[NSUB * DQ];                    // 64 KB
    const f4* b4 = (const f4*)base;
    for (int i = threadIdx.x; i < NSUB * DQ; i += blockDim.x) sB[i] = b4[i];
    if (blockIdx.x == 0 && threadIdx.x == 0) *maxslot = 0u;  // init for kernel 2
    __syncthreads();

    const f4* f4p  = (const f4*)feat;
    const int lane = threadIdx.x & 31;
    const int wave = (int)((blockIdx.x * blockDim.x + threadIdx.x) >> 5);
    const int nwav = (int)((gridDim.x * blockDim.x) >> 5);

    for (int row = wave; row < nrows; row += nwav) {
        int  s  = (int)sid[row];                    // 0..63
        long rb = (long)row * DQ;
        // 512B per wave per load, coalesced; NT so the one-shot stream
        // doesn't evict the resident baseline/L2 working set.
        f4 a0 = __builtin_nontemporal_load(&f4p[rb + lane]);
        f4 a1 = __builtin_nontemporal_load(&f4p[rb + lane + 32]);
        f4 c0 = sB[s * DQ + lane];
        f4 c1 = sB[s * DQ + lane + 32];
        f4 d0 = a0 - c0, d1 = a1 - c1;
        float acc = d0.x*d0.x + d0.y*d0.y + d0.z*d0.z + d0.w*d0.w
                  + d1.x*d1.x + d1.y*d1.y + d1.z*d1.z + d1.w*d1.w;
        // wave32 butterfly reduction
        #pragma unroll
        for (int off = 16; off > 0; off >>= 1) acc += __shfl_xor(acc, off, 32);
        if (lane == 0) raw[row] = sqrtf(acc);
    }
}

// ---------------------------------------------------------------------------
// Kernel 2: 5-tap zero-padded centered moving average (np.convolve 'same'),
// plus global max via block-reduce + atomicMax (float-as-uint, values >= 0).
// ---------------------------------------------------------------------------
__global__ __launch_bounds__(256) void smooth_kernel(
    const float* __restrict__ raw, float* __restrict__ devout,
    unsigned* __restrict__ maxslot, int n)
{
    int i = blockIdx.x * blockDim.x + threadIdx.x;
    float v = 0.0f;
    if (i < n) {
        float s = raw[i];
        if (i >= 1)     s += raw[i - 1];
        if (i >= 2)     s += raw[i - 2];
        if (i + 1 < n)  s += raw[i + 1];
        if (i + 2 < n)  s += raw[i + 2];
        v = s * 0.2f;
        devout[i] = v;
    }
    __shared__ float red[256];
    red[threadIdx.x] = v;
    __syncthreads();
    #pragma unroll
    for (int off = 128; off > 0; off >>= 1) {
        if (threadIdx.x < off)
            red[threadIdx.x] = fmaxf(red[threadIdx.x], red[threadIdx.x + off]);
        __syncthreads();
    }
    if (threadIdx.x == 0) atomicMax(maxslot, __float_as_uint(red[0]));
}

// ---------------------------------------------------------------------------
// Kernel 3: logits = relu(dev*W1 + b1) @ W2 + b2 via V_WMMA_F32_16X16X4_F32.
// One wave handles 16 samples (M=16). K=32 hidden units done as 8 WMMAs of
// K=4, accumulating into C seeded with the b2 column broadcast.
//   A frag (16x4 f32): lane m=lane&15 holds h[sample m][K = kk*4+khalf*2+v]
//   B frag (4x16 f32): lane n=lane&15 holds W2[K][n] (n>=3 -> 0)
//   D (16x16 f32):     VGPR r, lane-half -> rows r / r+8; cols 0..2 stored.
// W2 is staged zero-padded in LDS so the K-loop has NO divergence and no
// exec-masked loads: per iter = 4x ds_load_b32 + pk_fma + max + v_wmma.
// Also emits scores = dev / max.
// ---------------------------------------------------------------------------
__global__ __launch_bounds__(256) void mlp_wmma_kernel(
    const float* __restrict__ devs, const float* __restrict__ W1,
    const float* __restrict__ b1,   const float* __restrict__ W2,
    const float* __restrict__ b2,   const unsigned* __restrict__ maxslot,
    float* __restrict__ logits, float* __restrict__ scores, int n)
{
    __shared__ float sW2[32 * 16];   // [K=32][Ncol=16], cols 3..15 zero
    __shared__ float sW1[32];
    __shared__ float sb1[32];
    __shared__ float sb2[16];        // b2 padded with zeros

    const int t = threadIdx.x;
    sW2[t]       = 0.0f;             // zero the 512-entry tile (256 thr x2)
    sW2[t + 256] = 0.0f;
    if (t < 32) { sW1[t] = W1[t]; sb1[t] = b1[t]; }
    if (t < 16) { sb2[t] = (t < 3) ? b2[t] : 0.0f; }
    __syncthreads();                 // zeros visible before scatter-fill
    if (t < 96) {                    // W2[32,3] row-major -> padded [32][16]
        int j = t / 3, cidx = t % 3;
        sW2[j * 16 + cidx] = W2[t];
    }
    __syncthreads();

    const int lane  = t & 31;
    const int gw    = (int)((blockIdx.x * blockDim.x + t) >> 5);
    const int base  = gw << 4;
    if (base + 16 > n) return;                 // wave-uniform: EXEC stays full

    const int nn    = lane & 15;               // A: sample-in-tile, B/D: column
    const int khalf = lane >> 4;               // K half selector

    const float dv  = devs[base + nn];
    const float b2n = sb2[nn];
    v8f c = { b2n, b2n, b2n, b2n, b2n, b2n, b2n, b2n };

    #pragma unroll
    for (int kk = 0; kk < 8; ++kk) {
        const int j0 = kk * 4 + khalf * 2;     // K index for v=0 (v=1 -> j0+1)
        v2f a, bf;
        a.x  = fmaxf(fmaf(dv, sW1[j0],     sb1[j0]),     0.0f);
        a.y  = fmaxf(fmaf(dv, sW1[j0 + 1], sb1[j0 + 1]), 0.0f);
        bf.x = sW2[j0 * 16 + nn];
        bf.y = sW2[(j0 + 1) * 16 + nn];
        c = __builtin_amdgcn_wmma_f32_16x16x4_f32(
                /*neg_a=*/false, a, /*neg_b=*/false, bf,
                /*c_mod=*/(short)0, c, /*reuse_a=*/false, /*reuse_b=*/false);
    }

    if (nn < 3) {
        const int mbase = base + khalf * 8;
        #pragma unroll
        for (int r = 0; r < 8; ++r)
            logits[(size_t)(mbase + r) * 3 + nn] = c[r];
    }

    const float mx  = __uint_as_float(*maxslot);
    const float den = (mx > 0.0f) ? mx : 1.0f;
    if (khalf == 0) scores[base + lane] = dv / den;
}

// ---------------------------------------------------------------------------
extern "C" void kernel_launch(void* const* d_in, const int* in_sizes, int n_in,
                              void* d_out, int out_size, void* d_ws, size_t ws_size,
                              hipStream_t stream)
{
    const float*     feat = (const float*)d_in[0];
    const long long* sid  = (const long long*)d_in[1];
    const float*     base = (const float*)d_in[2];
    const float*     W1   = (const float*)d_in[3];
    const float*     b1   = (const float*)d_in[4];
    const float*     W2   = (const float*)d_in[5];
    const float*     b2   = (const float*)d_in[6];

    const int N = in_sizes[0] / 256;           // 524288

    float* out    = (float*)d_out;
    float* logits = out;                       // [N,3]
    float* devout = out + (size_t)3 * N;       // [N]
    float* scores = out + (size_t)4 * N;       // [N]

    float*    raw     = (float*)d_ws;                                  // [N]
    unsigned* maxslot = (unsigned*)((char*)d_ws + (size_t)N * sizeof(float));

    // 1) deviation norms (streams 512MB; 16384 waves, ~32 rows each)
    dev_norm_kernel<<<2048, 256, 0, stream>>>(feat, sid, base, raw, maxslot, N);

    // 2) smoothing + global max
    const int blocksB = (N + 255) / 256;
    smooth_kernel<<<blocksB, 256, 0, stream>>>(raw, devout, maxslot, N);

    // 3) WMMA MLP + scores (16 samples per wave, 8 waves per block)
    const int tiles   = N / 16;
    const int blocksC = (tiles + 7) / 8;
    mlp_wmma_kernel<<<blocksC, 256, 0, stream>>>(devout, W1, b1, W2, b2,
                                                 maxslot, logits, scores, N);
}